// SlotAttention_86887188398355
// MI455X (gfx1250) — compile-verified
//
#include <hip/hip_runtime.h>
#include <hip/hip_bf16.h>

typedef __attribute__((ext_vector_type(2))) float v2f;
typedef __attribute__((ext_vector_type(8))) float v8f;

#define BATCH 64
#define NTOK  4096
#define DIN   128
#define DSLOT 64
#define KSLOT 4
#define HID   128
#define NSPLIT 16           // blocks per batch in attention kernel
#define XPITCH 132          // padded LDS pitch for x_ln (132 % 64 == 4 -> conflict-free col reads)

__device__ __forceinline__ float sigm(float x) { return 1.0f / (1.0f + __expf(-x)); }

// ---------------------------------------------------------------------------
// Kernel A: fused LayerNorm(inputs) + k = x@Wk^T, v = x@Wv^T   (fp32 WMMA)
// block = 256 threads (8 waves), 128 rows per block, grid = B*N/128 = 2048
// Weights staged K-pair-interleaved: wp[p][j][e] = W[2p+e][j]  (one ds_load_b64
// per B-fragment, no VGPR shuffling before v_wmma).
// ---------------------------------------------------------------------------
__global__ __launch_bounds__(256) void kv_proj_kernel(
    const float* __restrict__ inputs,
    const float* __restrict__ ln_g, const float* __restrict__ ln_b,
    const float* __restrict__ Wk,   const float* __restrict__ Wv,
    float* __restrict__ kbuf, float* __restrict__ vbuf)
{
    __shared__ float xln[128][XPITCH];   // 128 rows of normalized x
    __shared__ float wp[64][128][2];     // K-pair interleaved fused [Wk|Wv]
    __shared__ float glds[128], blds[128];

    const int tid = threadIdx.x;

    // stage fused weight matrix [Wk | Wv], pair-interleaved over the c (K) dim
    for (int idx = tid; idx < 128 * 128; idx += 256) {
        int c = idx >> 7, j = idx & 127;
        float w = (j < 64) ? Wk[j * 128 + c] : Wv[(j - 64) * 128 + c];
        wp[c >> 1][j][c & 1] = w;
    }
    if (tid < 128) { glds[tid] = ln_g[tid]; blds[tid] = ln_b[tid]; }

    // LayerNorm: 2 threads per row, 64 contiguous elements each
    const int row  = tid >> 1;
    const int half = tid & 1;
    const long grow = (long)blockIdx.x * 128 + row;      // global row in [0, B*N)
    const float* xr = inputs + grow * DIN + half * 64;
    float s = 0.f, sq = 0.f;
    for (int i = 0; i < 64; ++i) {
        float v = xr[i];
        xln[row][half * 64 + i] = v;
        s += v; sq += v * v;
    }
    s  += __shfl_xor(s, 1);
    sq += __shfl_xor(sq, 1);
    float mean = s * (1.0f / 128.0f);
    float var  = sq * (1.0f / 128.0f) - mean * mean;
    float rstd = rsqrtf(var + 1e-5f);
    __syncthreads();                       // glds/blds + raw x staged
    for (int i = 0; i < 64; ++i) {
        int c = half * 64 + i;
        float v = xln[row][c];
        xln[row][c] = (v - mean) * rstd * glds[c] + blds[c];
    }
    __syncthreads();

    // GEMM: wave w owns rows [w*16, w*16+16), all 8 col-tiles (k:4, v:4)
    const int lane = tid & 31;
    const int wv   = tid >> 5;
    const int m    = lane & 15;       // row within tile (A) / col within tile (B,C)
    const int kh   = lane >> 4;       // K-half selector
    const int arow = wv * 16 + m;

    v8f acc[8] = {};
    for (int kk = 0; kk < 128; kk += 4) {
        v2f a = *(const v2f*)&xln[arow][kk + 2 * kh];     // aligned ds_load_b64
        const int p = (kk >> 1) + kh;                     // K-pair index
#pragma unroll
        for (int dt = 0; dt < 8; ++dt) {
            v2f b = *(const v2f*)&wp[p][dt * 16 + m][0];  // aligned ds_load_b64
            acc[dt] = __builtin_amdgcn_wmma_f32_16x16x4_f32(
                false, a, false, b, (short)0, acc[dt], false, false);
        }
    }

    // store C: lane<16 -> cols, vgpr i -> row i (+8 for hi lanes)
    const long rowBase = (long)blockIdx.x * 128 + wv * 16;
#pragma unroll
    for (int dt = 0; dt < 8; ++dt) {
#pragma unroll
        for (int i = 0; i < 8; ++i) {
            long r = rowBase + (i + kh * 8);
            long bb = r >> 12;          // / 4096
            long nn = r & 4095;
            int  j  = dt * 16 + m;
            float val = acc[dt][i];
            if (j < 64) kbuf[(bb * NTOK + nn) * DSLOT + j]        = val;
            else        vbuf[(bb * NTOK + nn) * DSLOT + (j - 64)] = val;
        }
    }
}

// ---------------------------------------------------------------------------
// Kernel B0: q = LN(slots) @ Wq^T.  grid = B blocks, 256 thr (k = tid/64, c = tid%64)
// ---------------------------------------------------------------------------
__global__ __launch_bounds__(256) void qproj_kernel(
    const float* __restrict__ slots,
    const float* __restrict__ ln_g, const float* __restrict__ ln_b,
    const float* __restrict__ Wq, float* __restrict__ qbuf)
{
    __shared__ float sn[KSLOT][DSLOT];
    const int tid = threadIdx.x;
    const int b = blockIdx.x, k = tid >> 6, c = tid & 63;
    float v = slots[((long)b * KSLOT + k) * DSLOT + c];
    sn[k][c] = v;
    __syncthreads();
    float s = 0.f, sq = 0.f;
    for (int i = 0; i < 64; ++i) { float t = sn[k][i]; s += t; sq += t * t; }
    float mean = s * (1.0f / 64.0f);
    float var  = sq * (1.0f / 64.0f) - mean * mean;
    float rstd = rsqrtf(var + 1e-5f);
    float nv = (v - mean) * rstd * ln_g[c] + ln_b[c];
    __syncthreads();
    sn[k][c] = nv;
    __syncthreads();
    float q = 0.f;
    const float* wr = Wq + c * 64;          // row j = c of Wq
    for (int i = 0; i < 64; ++i) q += sn[k][i] * wr[i];
    qbuf[((long)b * KSLOT + k) * DSLOT + c] = q;
}

// ---------------------------------------------------------------------------
// Kernel B1: logits -> softmax over K (local!) -> accumulate attn@v (WMMA) and
// denom = sum_n attn.  grid = B*NSPLIT blocks, 256 thr (8 waves, 32 rows/wave)
// attn stash is slot-major so A-fragments are one aligned ds_load_b64.
// ---------------------------------------------------------------------------
__global__ __launch_bounds__(256) void attn_kernel(
    const float* __restrict__ qbuf, const float* __restrict__ kbuf,
    const float* __restrict__ vbuf,
    float* __restrict__ updAcc, float* __restrict__ denomAcc,
    float* __restrict__ attnOut, int writeAttn)
{
    __shared__ float qlds[KSLOT][DSLOT];
    __shared__ float attnT[8][KSLOT][16];    // [wave][slot][token]  (slot-major)
    __shared__ float updlds[KSLOT][DSLOT];
    __shared__ float denlds[KSLOT];

    const int tid  = threadIdx.x;
    const int lane = tid & 31, wv = tid >> 5;
    const int b    = blockIdx.x >> 4;        // NSPLIT = 16
    const int nBase = (blockIdx.x & 15) * 256;

    ((float*)qlds)[tid]   = qbuf[(long)b * (KSLOT * DSLOT) + tid];
    ((float*)updlds)[tid] = 0.0f;
    if (tid < KSLOT) denlds[tid] = 0.0f;
    __syncthreads();

    const int half = lane >> 4;       // 0: dims 0..31, 1: dims 32..63  (logits phase)
    const int r    = lane & 15;       // token row within tile / slot idx (wmma phase)
    float denacc[KSLOT] = {0.f, 0.f, 0.f, 0.f};
    v8f acc[4] = {};

    for (int tile = 0; tile < 2; ++tile) {
        const int n0   = nBase + wv * 32 + tile * 16;
        const int nrow = n0 + r;
        const float* kp = kbuf + ((long)b * NTOK + nrow) * DSLOT + half * 32;

        float p0 = 0.f, p1 = 0.f, p2 = 0.f, p3 = 0.f;
        for (int i = 0; i < 32; ++i) {
            float kvv = kp[i];
            int c = half * 32 + i;
            p0 += qlds[0][c] * kvv;
            p1 += qlds[1][c] * kvv;
            p2 += qlds[2][c] * kvv;
            p3 += qlds[3][c] * kvv;
        }
        p0 += __shfl_xor(p0, 16); p1 += __shfl_xor(p1, 16);
        p2 += __shfl_xor(p2, 16); p3 += __shfl_xor(p3, 16);
        const float scale = 0.125f;                // SLOT_DIM^-0.5
        float l0 = p0 * scale, l1 = p1 * scale, l2 = p2 * scale, l3 = p3 * scale;
        float mx = fmaxf(fmaxf(l0, l1), fmaxf(l2, l3));
        float e0 = __expf(l0 - mx), e1 = __expf(l1 - mx);
        float e2 = __expf(l2 - mx), e3 = __expf(l3 - mx);
        float inv = 1.0f / (e0 + e1 + e2 + e3);
        float a0 = e0 * inv, a1 = e1 * inv, a2 = e2 * inv, a3 = e3 * inv;

        if (half == 0) {
            if (writeAttn) {
                attnOut[((long)b * KSLOT + 0) * NTOK + nrow] = a0;
                attnOut[((long)b * KSLOT + 1) * NTOK + nrow] = a1;
                attnOut[((long)b * KSLOT + 2) * NTOK + nrow] = a2;
                attnOut[((long)b * KSLOT + 3) * NTOK + nrow] = a3;
            }
            denacc[0] += a0; denacc[1] += a1; denacc[2] += a2; denacc[3] += a3;
            attnT[wv][0][r] = a0; attnT[wv][1][r] = a1;
            attnT[wv][2][r] = a2; attnT[wv][3][r] = a3;
        }
        __syncthreads();

        // updates[k,d] += attn[k,n] * v[n,d]  via  WMMA f32 16x16x4 (M=slot, K=n)
#pragma unroll
        for (int s = 0; s < 4; ++s) {
            const int rr = s * 4 + 2 * half;
            v2f a;
            if (r < KSLOT) a = *(const v2f*)&attnT[wv][r][rr];   // ds_load_b64
            else           { a.x = 0.0f; a.y = 0.0f; }
            const float* vp0 = vbuf + ((long)b * NTOK + n0 + rr) * DSLOT;
            const float* vp1 = vp0 + DSLOT;
#pragma unroll
            for (int dt = 0; dt < 4; ++dt) {
                v2f bb;
                bb.x = vp0[dt * 16 + r];
                bb.y = vp1[dt * 16 + r];
                acc[dt] = __builtin_amdgcn_wmma_f32_16x16x4_f32(
                    false, a, false, bb, (short)0, acc[dt], false, false);
            }
        }
        __syncthreads();
    }

    // reduce: only lanes<16, vgprs 0..3 carry valid slots 0..3
    if (half == 0) {
#pragma unroll
        for (int dt = 0; dt < 4; ++dt)
#pragma unroll
            for (int i = 0; i < KSLOT; ++i)
                atomicAdd(&updlds[i][dt * 16 + r], acc[dt][i]);
#pragma unroll
        for (int k = 0; k < KSLOT; ++k) atomicAdd(&denlds[k], denacc[k]);
    }
    __syncthreads();
    atomicAdd(&updAcc[(long)b * (KSLOT * DSLOT) + tid], ((float*)updlds)[tid]);
    if (tid < KSLOT) atomicAdd(&denomAcc[(long)b * KSLOT + tid], denlds[tid]);
}

// ---------------------------------------------------------------------------
// Kernel B2: normalize updates, GRUCell, LN, MLP residual.
// grid = 32 blocks x 8 waves = 256 waves; one wave per (b,k) slot row.
// ---------------------------------------------------------------------------
__global__ __launch_bounds__(256) void gru_mlp_kernel(
    const float* __restrict__ updAcc, const float* __restrict__ denomAcc,
    const float* __restrict__ slotsIn,
    const float* __restrict__ Wih, const float* __restrict__ Whh,
    const float* __restrict__ bih, const float* __restrict__ bhh,
    const float* __restrict__ lnm_g, const float* __restrict__ lnm_b,
    const float* __restrict__ W1, const float* __restrict__ b1,
    const float* __restrict__ W2, const float* __restrict__ b2,
    float* __restrict__ slotsOut)
{
    __shared__ float xs[8][64], hs[8][64], ms[8][64], h1s[8][128];
    const int tid = threadIdx.x, lane = tid & 31, wv = tid >> 5;
    const int gid = blockIdx.x * 8 + wv;          // (b*4 + k)
    const long base = (long)gid * DSLOT;

    float den = denomAcc[gid] + 1e-8f;
    float x0 = updAcc[base + lane] / den;
    float x1 = updAcc[base + lane + 32] / den;
    float h0 = slotsIn[base + lane];
    float h1v = slotsIn[base + lane + 32];
    xs[wv][lane] = x0; xs[wv][lane + 32] = x1;
    hs[wv][lane] = h0; hs[wv][lane + 32] = h1v;
    __syncthreads();

    // gates: rows [0,64)=r, [64,128)=z, [128,192)=n of W_ih/W_hh
    float gi[6], gh[6];
#pragma unroll
    for (int t = 0; t < 6; ++t) {
        int j = lane + (t & 1) * 32 + (t >> 1) * 64;
        float si = bih[j], sh = bhh[j];
        const float* wi = Wih + j * 64;
        const float* wh = Whh + j * 64;
        for (int c = 0; c < 64; ++c) {
            si += wi[c] * xs[wv][c];
            sh += wh[c] * hs[wv][c];
        }
        gi[t] = si; gh[t] = sh;
    }
    float r0 = sigm(gi[0] + gh[0]), r1 = sigm(gi[1] + gh[1]);
    float z0 = sigm(gi[2] + gh[2]), z1 = sigm(gi[3] + gh[3]);
    float n0 = tanhf(gi[4] + r0 * gh[4]), n1 = tanhf(gi[5] + r1 * gh[5]);
    float s0 = (1.f - z0) * n0 + z0 * h0;
    float s1 = (1.f - z1) * n1 + z1 * h1v;

    // LayerNorm over 64 dims (2 per lane, butterfly reduce over 32 lanes)
    float ss = s0 + s1, sq = s0 * s0 + s1 * s1;
    for (int o = 1; o < 32; o <<= 1) { ss += __shfl_xor(ss, o); sq += __shfl_xor(sq, o); }
    float mean = ss * (1.0f / 64.0f);
    float var  = sq * (1.0f / 64.0f) - mean * mean;
    float rstd = rsqrtf(var + 1e-5f);
    float m0 = (s0 - mean) * rstd * lnm_g[lane] + lnm_b[lane];
    float m1 = (s1 - mean) * rstd * lnm_g[lane + 32] + lnm_b[lane + 32];
    ms[wv][lane] = m0; ms[wv][lane + 32] = m1;
    __syncthreads();

    // h1 = relu(m @ W1^T + b1),  128 outputs, 4 per lane
#pragma unroll
    for (int t = 0; t < 4; ++t) {
        int j = lane + t * 32;
        float a = b1[j];
        const float* w = W1 + j * 64;
        for (int c = 0; c < 64; ++c) a += w[c] * ms[wv][c];
        h1s[wv][j] = fmaxf(a, 0.0f);
    }
    __syncthreads();

    // out = slots + h1 @ W2^T + b2
    float o0 = b2[lane], o1 = b2[lane + 32];
    const float* w20 = W2 + lane * 128;
    const float* w21 = W2 + (lane + 32) * 128;
    for (int c = 0; c < 128; ++c) {
        float hc = h1s[wv][c];
        o0 += w20[c] * hc;
        o1 += w21[c] * hc;
    }
    slotsOut[base + lane]      = s0 + o0;
    slotsOut[base + lane + 32] = s1 + o1;
}

// ---------------------------------------------------------------------------
extern "C" void kernel_launch(void* const* d_in, const int* in_sizes, int n_in,
                              void* d_out, int out_size, void* d_ws, size_t ws_size,
                              hipStream_t stream)
{
    const float* inputs  = (const float*)d_in[0];
    const float* slots0  = (const float*)d_in[1];
    const float* ln_in_g = (const float*)d_in[2];
    const float* ln_in_b = (const float*)d_in[3];
    const float* ln_s_g  = (const float*)d_in[4];
    const float* ln_s_b  = (const float*)d_in[5];
    const float* ln_m_g  = (const float*)d_in[6];
    const float* ln_m_b  = (const float*)d_in[7];
    const float* Wq      = (const float*)d_in[8];
    const float* Wk      = (const float*)d_in[9];
    const float* Wv      = (const float*)d_in[10];
    const float* Wih     = (const float*)d_in[11];
    const float* Whh     = (const float*)d_in[12];
    const float* bih     = (const float*)d_in[13];
    const float* bhh     = (const float*)d_in[14];
    const float* W1      = (const float*)d_in[15];
    const float* b1      = (const float*)d_in[16];
    const float* W2      = (const float*)d_in[17];
    const float* b2      = (const float*)d_in[18];

    float* out = (float*)d_out;
    float* ws  = (float*)d_ws;

    const size_t KV = (size_t)BATCH * NTOK * DSLOT;   // 16,777,216 floats each
    const size_t SL = (size_t)BATCH * KSLOT * DSLOT;  // 16,384 floats
    float* kbuf     = ws;
    float* vbuf     = kbuf + KV;
    float* qbuf     = vbuf + KV;
    float* slotsbuf = qbuf + SL;
    float* updAcc   = slotsbuf + SL;
    float* denomAcc = updAcc + SL;                    // contiguous with updAcc

    hipMemcpyAsync(slotsbuf, slots0, SL * sizeof(float),
                   hipMemcpyDeviceToDevice, stream);

    kv_proj_kernel<<<(BATCH * NTOK) / 128, 256, 0, stream>>>(
        inputs, ln_in_g, ln_in_b, Wk, Wv, kbuf, vbuf);

    float* attnOut = out + SL;   // d_out = [slots | attn_weights]

    for (int it = 0; it < 3; ++it) {
        qproj_kernel<<<BATCH, 256, 0, stream>>>(slotsbuf, ln_s_g, ln_s_b, Wq, qbuf);
        hipMemsetAsync(updAcc, 0, (SL + BATCH * KSLOT) * sizeof(float), stream);
        attn_kernel<<<BATCH * NSPLIT, 256, 0, stream>>>(
            qbuf, kbuf, vbuf, updAcc, denomAcc, attnOut, (it == 2) ? 1 : 0);
        gru_mlp_kernel<<<32, 256, 0, stream>>>(
            updAcc, denomAcc, slotsbuf, Wih, Whh, bih, bhh,
            ln_m_g, ln_m_b, W1, b1, W2, b2,
            (it < 2) ? slotsbuf : out);
    }
}